// AdaptiveLIF_53025666236691
// MI455X (gfx1250) — compile-verified
//
#include <hip/hip_runtime.h>
#include <math.h>

typedef __attribute__((ext_vector_type(2))) float v2f;
typedef __attribute__((ext_vector_type(8))) float v8f;
typedef __attribute__((ext_vector_type(4))) int   v4i;

#define T_ 16
#define B_ 16
#define C_ 8
#define H_ 128
#define W_ 128
#define HW_ (H_ * W_)     /* 16384 */
#define CHW_ (C_ * HW_)   /* 131072 */

#define TILE_ROWS 32
#define NTILES (H_ / TILE_ROWS) /* 4 */
#define TILE_ELEMS ((TILE_ROWS + 2) * W_)

// --- gfx1250 async global->LDS support (guarded; fallback = plain LDS fill) --
#if defined(__has_builtin)
#if __has_builtin(__builtin_amdgcn_global_load_async_to_lds_b128)
#define HAVE_ASYNC_LDS 1
#endif
#endif

#ifdef HAVE_ASYNC_LDS
typedef __attribute__((address_space(1))) v4i gv4i_t;   // global 16B chunk
typedef __attribute__((address_space(3))) v4i lv4i_t;   // LDS 16B chunk
__device__ __forceinline__ void async_wait0() {
#if __has_builtin(__builtin_amdgcn_s_wait_asynccnt)
    __builtin_amdgcn_s_wait_asynccnt(0);
#else
    asm volatile("s_wait_asynccnt 0x0" ::: "memory");
#endif
}
#endif

// ---------------------------------------------------------------------------
// Kernel 1: per-(t,b) statistics.
//   - channel max / avg  -> pooled[tb][16]  (concat [max, avg], ref order)
//   - density, temporal |diff|, motion (unfold-variance mean) -> thr[tb]
// One workgroup per (t,b). Channel-sum plane lives in LDS so the 3x3 box
// filter (s1) can be squared+summed; s2's pixel sum collapses to a
// border-count weighted sum of per-pixel sum-of-squares.
// ---------------------------------------------------------------------------
__global__ __launch_bounds__(256)
void stats_kernel(const float* __restrict__ events,
                  const float* __restrict__ density_w,
                  const float* __restrict__ temporal_w,
                  const float* __restrict__ motion_w,
                  float* __restrict__ ws_thr,      // [256]
                  float* __restrict__ ws_pooled)   // [256][16]
{
    __shared__ float xsum[HW_];
    __shared__ float red[256];

    const int tb  = blockIdx.x;       // t*B + b
    const int t   = tb / B_;
    const int tid = threadIdx.x;
    const float* x  = events + (size_t)tb * CHW_;
    const float* xp = events + (size_t)(t > 0 ? tb - B_ : tb) * CHW_;

    float cmax[C_], csum[C_];
#pragma unroll
    for (int c = 0; c < C_; ++c) { cmax[c] = -3.402823466e38f; csum[c] = 0.f; }
    float wsq = 0.f, tdiff = 0.f;

    for (int p = tid; p < HW_; p += 256) {
        const int py = p >> 7, px = p & 127;
        const float cnt = ((py == 0 || py == H_ - 1) ? 2.f : 3.f) *
                          ((px == 0 || px == W_ - 1) ? 2.f : 3.f);
        float s = 0.f, sq = 0.f;
#pragma unroll
        for (int c = 0; c < C_; ++c) {
            const float vch = x[c * HW_ + p];
            cmax[c] = fmaxf(cmax[c], vch);
            csum[c] += vch;
            s  += vch;
            sq += vch * vch;
            if (t > 0) tdiff += fabsf(vch - xp[c * HW_ + p]);
        }
        xsum[p] = s;
        wsq += cnt * sq;                 // == sum over pixels of s2(p)
    }
    __syncthreads();

    // sum over pixels of s1(p)^2, s1 = zero-padded 3x3 box sum of xsum
    float s1sq = 0.f;
    for (int p = tid; p < HW_; p += 256) {
        const int py = p >> 7, px = p & 127;
        float s1 = 0.f;
#pragma unroll
        for (int dy = -1; dy <= 1; ++dy) {
            const int yy = py + dy;
            if (yy < 0 || yy >= H_) continue;
#pragma unroll
            for (int dx = -1; dx <= 1; ++dx) {
                const int xx = px + dx;
                if (xx < 0 || xx >= W_) continue;
                s1 += xsum[yy * W_ + xx];
            }
        }
        s1sq += s1 * s1;
    }

    auto reduceSum = [&](float v) -> float {
        red[tid] = v; __syncthreads();
        for (int s = 128; s > 0; s >>= 1) {
            if (tid < s) red[tid] += red[tid + s];
            __syncthreads();
        }
        const float r = red[0]; __syncthreads();
        return r;
    };
    auto reduceMax = [&](float v) -> float {
        red[tid] = v; __syncthreads();
        for (int s = 128; s > 0; s >>= 1) {
            if (tid < s) red[tid] = fmaxf(red[tid], red[tid + s]);
            __syncthreads();
        }
        const float r = red[0]; __syncthreads();
        return r;
    };

    float totsum = 0.f;
#pragma unroll
    for (int c = 0; c < C_; ++c) {
        const float m = reduceMax(cmax[c]);
        const float s = reduceSum(csum[c]);
        if (tid == 0) {
            ws_pooled[tb * 16 + c]     = m;                 // max pool
            ws_pooled[tb * 16 + 8 + c] = s * (1.f / HW_);   // avg pool
        }
        totsum += s;
    }
    const float WSQ  = reduceSum(wsq);
    const float S1SQ = reduceSum(s1sq);
    const float TD   = reduceSum(tdiff);

    if (tid == 0) {
        const float density  = totsum * (1.f / CHW_);
        const float temporal = (t > 0) ? TD * (1.f / CHW_) : 0.f;
        const float n = (float)(C_ * 9);                    // 72
        const float motion = (WSQ - S1SQ / n) / ((n - 1.f) * (float)HW_);
        const float z = density_w[0] * density + temporal_w[0] * temporal +
                        motion_w[0] * motion;
        const float adj = 2.f / (1.f + expf(-z));
        ws_thr[tb] = 1.f + adj;                             // base_threshold 1.0
    }
}

// ---------------------------------------------------------------------------
// Kernel 2: beta via WMMA.  h = relu(pooled(256x16) @ w1^T(16x1)) done with
// V_WMMA_F32_16X16X4_F32: 16 waves, wave w owns rows 16w..16w+15, looping
// four K=4 chunks.  B columns replicate w1, so every D column equals h.
// A layout (32b 16x4): lane<16 -> {K=k0,k0+1}, lane>=16 -> {K=k0+2,k0+3}.
// D layout (32b 16x16): VGPR r: lanes0-15 M=r, lanes16-31 M=r+8.
// ---------------------------------------------------------------------------
__global__ __launch_bounds__(512)
void attn_kernel(const float* __restrict__ ws_pooled, // [256][16]
                 const float* __restrict__ att_w1,    // [1][16]
                 const float* __restrict__ att_w2,    // [8][1]
                 float* __restrict__ ws_beta)         // [256][8]
{
    __shared__ float hbuf[256];
    const int tid   = threadIdx.x;
    const int wave  = tid >> 5;
    const int lane  = tid & 31;
    const int base  = wave * 16;
    const int row   = base + (lane & 15);
    const int khalf = (lane >> 4) * 2;   // 0 for lanes<16, 2 for lanes>=16

    v8f acc = {};
#pragma unroll
    for (int k0 = 0; k0 < 16; k0 += 4) {
        v2f a, b;
        a.x = ws_pooled[row * 16 + k0 + khalf + 0];
        a.y = ws_pooled[row * 16 + k0 + khalf + 1];
        b.x = att_w1[k0 + khalf + 0];    // broadcast across all N columns
        b.y = att_w1[k0 + khalf + 1];
        acc = __builtin_amdgcn_wmma_f32_16x16x4_f32(
            /*neg_a=*/false, a, /*neg_b=*/false, b,
            /*c_mod=*/(short)0, acc, /*reuse_a=*/false, /*reuse_b=*/false);
    }

    if (lane == 0) {
#pragma unroll
        for (int r = 0; r < 8; ++r) hbuf[base + r] = acc[r];
    } else if (lane == 16) {
#pragma unroll
        for (int r = 0; r < 8; ++r) hbuf[base + 8 + r] = acc[r];
    }
    __syncthreads();

#pragma unroll
    for (int j = 0; j < 4; ++j) {
        const int idx = tid * 4 + j;         // 0..2047 == m*8 + c
        const int m = idx >> 3, c = idx & 7;
        const float h = fmaxf(hbuf[m], 0.f); // relu, hidden == 1
        ws_beta[idx] = 0.5f + 0.45f / (1.f + expf(-h * att_w2[c]));
    }
}

// ---------------------------------------------------------------------------
// Kernel 3: depthwise 3x3 conv fused with the sequential LIF scan.
// Workgroup = (b, c, 32-row tile); loops t keeping membrane v in registers
// (16 pixels/thread).  Frame tiles staged global->LDS with gfx1250 async
// LDS loads (ASYNCcnt), DOUBLE-BUFFERED: frame t+1 streams into the other
// LDS buffer while timestep t computes, so the async copy overlaps VALU.
// OOB halo rows are zeroed once and never rewritten.  Output stored
// non-temporal so the 128 MB input stays resident in the 192 MB L2.
// ---------------------------------------------------------------------------
__global__ __launch_bounds__(256)
void lif_kernel(const float* __restrict__ events,
                const float* __restrict__ conv_w,   // [C][1][3][3]
                const float* __restrict__ ws_beta,  // [256][8]
                const float* __restrict__ ws_thr,   // [256]
                float* __restrict__ out)
{
    __shared__ __align__(16) float tile[2][TILE_ELEMS];
    const int tid     = threadIdx.x;
    const int tileIdx = blockIdx.x & (NTILES - 1);
    const int c       = (blockIdx.x >> 2) & (C_ - 1);
    const int b       = blockIdx.x >> 5;
    const int r0      = tileIdx * TILE_ROWS;

    // valid LDS rows to copy each step (halo rows outside the image excluded)
    const int lyStart = (tileIdx == 0) ? 1 : 0;
    const int lyEnd   = (tileIdx == NTILES - 1) ? TILE_ROWS : TILE_ROWS + 1;
    const int nChunks = (lyEnd - lyStart + 1) * (W_ / 4);  // float4 chunks

    // zero OOB halo rows once (both buffers); copies never touch them again
    if (tileIdx == 0)
        for (int i = tid; i < W_; i += 256) { tile[0][i] = 0.f; tile[1][i] = 0.f; }
    if (tileIdx == NTILES - 1)
        for (int i = tid; i < W_; i += 256) {
            tile[0][(TILE_ROWS + 1) * W_ + i] = 0.f;
            tile[1][(TILE_ROWS + 1) * W_ + i] = 0.f;
        }

    auto issueLoad = [&](int t, int bufIdx) {
        const float* xin = events + ((size_t)(t * B_ + b) * C_ + c) * HW_;
        float* dst = tile[bufIdx];
        for (int idx = tid; idx < nChunks; idx += 256) {
            const int ly   = lyStart + (idx >> 5);   // W_/4 == 32 chunks/row
            const int colc = (idx & 31) * 4;
            const int gy   = r0 + ly - 1;            // in [0, H_) by design
#ifdef HAVE_ASYNC_LDS
            __builtin_amdgcn_global_load_async_to_lds_b128(
                (gv4i_t*)(xin + gy * W_ + colc),
                (lv4i_t*)&dst[ly * W_ + colc], 0, 0);
#else
            *(float4*)&dst[ly * W_ + colc] =
                *(const float4*)&xin[gy * W_ + colc];
#endif
        }
    };

    float w[9];
#pragma unroll
    for (int i = 0; i < 9; ++i) w[i] = conv_w[c * 9 + i];

    float v[16];
#pragma unroll
    for (int i = 0; i < 16; ++i) v[i] = 0.f;

    issueLoad(0, 0);                  // prime the pipeline
    int cur = 0;

    for (int t = 0; t < T_; ++t) {
#ifdef HAVE_ASYNC_LDS
        async_wait0();  // this wave's copies into tile[cur] have landed
#endif
        __syncthreads(); // all waves' copies visible; prior compute finished

        if (t + 1 < T_) issueLoad(t + 1, cur ^ 1);   // overlap next copy

        const size_t frame = ((size_t)(t * B_ + b) * C_ + c) * HW_;
        float* xout = out + frame;
        const float beta = ws_beta[(t * B_ + b) * C_ + c];
        const float thr  = ws_thr[t * B_ + b];
        const float* tl  = tile[cur];

#pragma unroll
        for (int i = 0; i < 16; ++i) {
            const int p  = tid + i * 256;       // 0..4095 within tile
            const int py = p >> 7;              // 0..31
            const int px = p & 127;
            const bool lOK = px > 0, rOK = px < W_ - 1;
            const float* rm = &tl[(py + 0) * W_ + px];
            const float* rc = &tl[(py + 1) * W_ + px];
            const float* rp = &tl[(py + 2) * W_ + px];

            float wi = 0.f;
            wi += w[0] * (lOK ? rm[-1] : 0.f);
            wi += w[1] * rm[0];
            wi += w[2] * (rOK ? rm[1] : 0.f);
            wi += w[3] * (lOK ? rc[-1] : 0.f);
            wi += w[4] * rc[0];
            wi += w[5] * (rOK ? rc[1] : 0.f);
            wi += w[6] * (lOK ? rp[-1] : 0.f);
            wi += w[7] * rp[0];
            wi += w[8] * (rOK ? rp[1] : 0.f);

            float vv    = beta * v[i] + (1.f - beta) * wi;        // charge
            float spike = atanf(2.f * (vv - thr)) * 0.5f + 0.5f;  // surrogate
            // NT store: output is write-once, keep events L2-resident instead
            __builtin_nontemporal_store(rc[0] * spike,
                                        &xout[(r0 + py) * W_ + px]);
            v[i] = (1.f - spike) * vv;                            // reset
        }
        cur ^= 1;
    }
}

// ---------------------------------------------------------------------------
extern "C" void kernel_launch(void* const* d_in, const int* in_sizes, int n_in,
                              void* d_out, int out_size, void* d_ws, size_t ws_size,
                              hipStream_t stream) {
    const float* events    = (const float*)d_in[0];
    const float* conv_w    = (const float*)d_in[1];
    const float* att_w1    = (const float*)d_in[2];
    const float* att_w2    = (const float*)d_in[3];
    const float* density_w = (const float*)d_in[4];
    const float* temporal_w= (const float*)d_in[5];
    const float* motion_w  = (const float*)d_in[6];
    float* out = (float*)d_out;

    float* ws        = (float*)d_ws;
    float* ws_thr    = ws;                 // 256
    float* ws_pooled = ws + 256;           // 256*16
    float* ws_beta   = ws + 256 + 4096;    // 256*8

    stats_kernel<<<T_ * B_, 256, 0, stream>>>(events, density_w, temporal_w,
                                              motion_w, ws_thr, ws_pooled);
    attn_kernel<<<1, 512, 0, stream>>>(ws_pooled, att_w1, att_w2, ws_beta);
    lif_kernel<<<B_ * C_ * NTILES, 256, 0, stream>>>(events, conv_w, ws_beta,
                                                     ws_thr, out);
}